// SoftProjection_24455543783470
// MI455X (gfx1250) — compile-verified
//
#include <hip/hip_runtime.h>
#include <math.h>

// ---------------------------------------------------------------------------
// SoftProjection for MI455X (gfx1250, wave32)
//   B=4, N=8192, M=2048, F=256, k=16
//   - KNN cross terms via V_WMMA_F32_16X16X4_F32 (16q x 16p tiles, K=4 pad 0)
//   - point tiles async-loaded straight into WMMA B-matrix layout in LDS,
//     double-buffered (issue tile i+1 while computing tile i)
//   - per-thread register top-16, softmax, gather-weighted sums
// ---------------------------------------------------------------------------

#define AS1 __attribute__((address_space(1)))
#define AS3 __attribute__((address_space(3)))

constexpr int Bn  = 4;
constexpr int Nn  = 8192;
constexpr int Mn  = 2048;
constexpr int Fn  = 256;
constexpr int KNN = 16;
constexpr int NT  = 64;   // points per LDS tile
constexpr int QT  = 64;   // queries per block (1 per thread)
constexpr int BT  = 64;   // threads per block (2 waves)
constexpr int NSUB = NT / 16;          // 16-point subtiles per tile
constexpr int NTILES = Nn / NT;

typedef __attribute__((ext_vector_type(2))) float v2f;
typedef __attribute__((ext_vector_type(8))) float v8f;

#if defined(__gfx1250__) && __has_builtin(__builtin_amdgcn_global_load_async_to_lds_b32)
#define HAVE_ASYNC 1
#else
#define HAVE_ASYNC 0
#endif

__global__ __launch_bounds__(BT) void softproj_kernel(
    const float* __restrict__ pc,    // (B,3,N)
    const float* __restrict__ qc,    // (B,3,M)
    const float* __restrict__ pf,    // (B,F,N)
    const float* __restrict__ temp,  // scalar
    float* __restrict__ out)         // (B,3+F,M)
{
  // B-matrix layout per subtile: entries 0..15 = {x,y} (K0,K1), 16..31 = {z,0} (K2,K3)
  __shared__ v2f   bmat[2][NSUB * 32];   // 2 x 1 KB
  __shared__ float p2s[2][NT];           // 2 x 256 B
  __shared__ float d2s[QT * NT];         // 16 KB

  const int t    = threadIdx.x;
  const int lane = t & 31;
  const int wave = t >> 5;
  const int b    = blockIdx.x / (Mn / QT);
  const int qb   = blockIdx.x % (Mn / QT);
  const int mbase = qb * QT;

  const float* PCb = pc + b * 3 * Nn;
  const float* QCb = qc + b * 3 * Mn;
  const float* PFb = pf + (size_t)b * Fn * Nn;
  float* outb = out + (size_t)b * (3 + Fn) * Mn;

  // this thread's staging slot: point (nb + t) -> subtile ts, column tl
  const int ts = t >> 4;
  const int tl = t & 15;

  // one-time: zero the K3 pad slots (never touched by the async loads)
  bmat[0][ts * 32 + 16 + tl].y = 0.0f;
  bmat[1][ts * 32 + 16 + tl].y = 0.0f;

  // stage tile starting at point nb into buffer buf (async scatter into B layout)
  auto stage_tile = [&](int nb, int buf) {
#if HAVE_ASYNC
    float* e01 = (float*)&bmat[buf][ts * 32 + tl];        // {x,y}
    float* e23 = (float*)&bmat[buf][ts * 32 + 16 + tl];   // {z,-}
    __builtin_amdgcn_global_load_async_to_lds_b32(
        (AS1 int*)(PCb + 0 * Nn + nb + t), (AS3 int*)(e01 + 0), 0, 0);
    __builtin_amdgcn_global_load_async_to_lds_b32(
        (AS1 int*)(PCb + 1 * Nn + nb + t), (AS3 int*)(e01 + 1), 0, 0);
    __builtin_amdgcn_global_load_async_to_lds_b32(
        (AS1 int*)(PCb + 2 * Nn + nb + t), (AS3 int*)(e23 + 0), 0, 0);
#else
    const float x = PCb[0 * Nn + nb + t];
    const float y = PCb[1 * Nn + nb + t];
    const float z = PCb[2 * Nn + nb + t];
    bmat[buf][ts * 32 + tl].x = x;
    bmat[buf][ts * 32 + tl].y = y;
    bmat[buf][ts * 32 + 16 + tl].x = z;
#endif
  };

  // ---- A-matrix (16x4, M=queries, K=coords[x,y,z,0]) per 16-row group ----
  // layout: lanes 0-15 hold M=lane, VGPR0=K0, VGPR1=K1; lanes 16-31 K2,K3.
  const int lm = lane & 15;
  const int hi = (lane >= 16) ? 1 : 0;
  v2f   aG[2];
  float q2v[2][8];
#pragma unroll
  for (int g = 0; g < 2; ++g) {
    const int m0 = mbase + wave * 32 + g * 16;
    if (!hi) {
      aG[g].x = QCb[0 * Mn + m0 + lm];
      aG[g].y = QCb[1 * Mn + m0 + lm];
    } else {
      aG[g].x = QCb[2 * Mn + m0 + lm];   // K=2 (z)
      aG[g].y = 0.0f;                    // K=3 pad
    }
#pragma unroll
    for (int r = 0; r < 8; ++r) {        // |q|^2 for the C rows this lane owns
      const int mq = m0 + r + hi * 8;
      const float qx = QCb[mq], qy = QCb[Mn + mq], qz = QCb[2 * Mn + mq];
      q2v[g][r] = qx * qx + qy * qy + qz * qz;
    }
  }

  float bestD[KNN];
  int   bestI[KNN];
#pragma unroll
  for (int k = 0; k < KNN; ++k) { bestD[k] = 3.4e38f; bestI[k] = 0; }

  stage_tile(0, 0);   // prologue

  for (int it = 0; it < NTILES; ++it) {
    const int cur = it & 1;
    const int nb  = it * NT;
#if HAVE_ASYNC
#if __has_builtin(__builtin_amdgcn_s_wait_asynccnt)
    __builtin_amdgcn_s_wait_asynccnt(0);
#else
    asm volatile("s_wait_asynccnt 0" ::: "memory");
#endif
#endif
    __syncthreads();   // buf[cur] landed everywhere; prev consume finished

    // |p|^2 for this tile: thread t reads back the 3 words it loaded itself
    {
      const float x = bmat[cur][ts * 32 + tl].x;
      const float y = bmat[cur][ts * 32 + tl].y;
      const float z = bmat[cur][ts * 32 + 16 + tl].x;
      p2s[cur][t] = x * x + y * y + z * z;
    }

    // prefetch next tile into the other buffer (its last reader was 2 barriers ago)
    if (it + 1 < NTILES) stage_tile(nb + NT, cur ^ 1);

    __syncthreads();   // p2s[cur] visible to all

    // ---- d2 tiles via WMMA: each wave does 2 row-groups x 4 subtiles ----
#pragma unroll
    for (int s = 0; s < NSUB; ++s) {
      const v2f bm = bmat[cur][s * 32 + lane];   // single b64, branch-free
      const float p2c = p2s[cur][s * 16 + lm];
#pragma unroll
      for (int g = 0; g < 2; ++g) {
        v8f c = {};
        c = __builtin_amdgcn_wmma_f32_16x16x4_f32(false, aG[g], false, bm,
                                                  (short)0, c, false, false);
        // C layout: VGPR r = row (rbase+r), lane = column
        const int rbase = wave * 32 + g * 16 + hi * 8;
        const int nl = s * 16 + lm;
#pragma unroll
        for (int r = 0; r < 8; ++r) {
          d2s[(rbase + r) * NT + nl] = q2v[g][r] + p2c - 2.0f * c[r];
        }
      }
    }
    __syncthreads();

    // ---- consume: thread t scans its query row, rotated (bank-conflict-free)
#pragma unroll 4
    for (int j = 0; j < NT; ++j) {
      const int col = (j + t) & (NT - 1);
      const float d = d2s[t * NT + col];
      if (d < bestD[KNN - 1]) {
        float dd = d; int ii = nb + col;
#pragma unroll
        for (int q = 0; q < KNN; ++q) {
          if (dd < bestD[q]) {
            const float td = bestD[q]; const int ti = bestI[q];
            bestD[q] = dd; bestI[q] = ii; dd = td; ii = ti;
          }
        }
      }
    }
  }

  // ---- softmax weights over the 16 neighbors ----
  const float tv = temp[0];
  const float sigma = fmaxf(tv * tv, 1e-4f);
  const float invSigma = 1.0f / sigma;
  const float dmin = bestD[0];   // sorted ascending
  float w[KNN];
  float wsum = 0.0f;
#pragma unroll
  for (int k = 0; k < KNN; ++k) {
    w[k] = __expf((dmin - bestD[k]) * invSigma);
    wsum += w[k];
  }
  const float invW = 1.0f / wsum;
#pragma unroll
  for (int k = 0; k < KNN; ++k) w[k] *= invW;

  const int m = mbase + t;

  // ---- projected points (3 channels) ----
  float sx = 0.f, sy = 0.f, sz = 0.f;
#pragma unroll
  for (int k = 0; k < KNN; ++k) {
    const int n = bestI[k];
    sx += w[k] * PCb[0 * Nn + n];
    sy += w[k] * PCb[1 * Nn + n];
    sz += w[k] * PCb[2 * Nn + n];
  }
  outb[0 * Mn + m] = sx;
  outb[1 * Mn + m] = sy;
  outb[2 * Mn + m] = sz;

  // ---- propagated features (F channels), coalesced stores over m ----
#pragma unroll 2
  for (int f = 0; f < Fn; ++f) {
    const float* row = PFb + (size_t)f * Nn;
    float acc = 0.f;
#pragma unroll
    for (int k = 0; k < KNN; ++k) acc += w[k] * row[bestI[k]];
    outb[(size_t)(3 + f) * Mn + m] = acc;
  }
}

extern "C" void kernel_launch(void* const* d_in, const int* in_sizes, int n_in,
                              void* d_out, int out_size, void* d_ws, size_t ws_size,
                              hipStream_t stream) {
  const float* pc   = (const float*)d_in[0];  // (4,3,8192)
  const float* qc   = (const float*)d_in[1];  // (4,3,2048)
  const float* pf   = (const float*)d_in[2];  // (4,256,8192)
  const float* temp = (const float*)d_in[3];  // scalar
  // d_in[4] = group_size (16), compile-time constant KNN here
  float* out = (float*)d_out;                 // (4,259,2048)

  const int grid = Bn * (Mn / QT);            // 128 blocks x 64 threads
  softproj_kernel<<<grid, BT, 0, stream>>>(pc, qc, pf, temp, out);
}